// GraphLayer_6811818131656
// MI455X (gfx1250) — compile-verified
//
#include <hip/hip_runtime.h>
#include <hip/hip_bf16.h>

#define N_CODES   4880
#define CODE_SIZE 28
#define GRAPH_SIZE 32
#define NX 112        // concat width: 56 center + 56 neigh
#define NZ 56         // width of Z rows

typedef __attribute__((ext_vector_type(2))) float v2f;
typedef __attribute__((ext_vector_type(8))) float v8f;

// ---------------------------------------------------------------------------
// Phase 1: build X[i][0:112] = [center_emb | ccode_cat | neigh_emb | ncat]
// One thread per (i, k) with k in [0,28).
// ---------------------------------------------------------------------------
__global__ void build_X(const float* __restrict__ code_x,
                        const float* __restrict__ neighbor,
                        const float* __restrict__ c_emb,
                        const float* __restrict__ n_emb,
                        const float* __restrict__ cat_table,
                        const int*   __restrict__ cat_idx,
                        float* __restrict__ X) {
    int idx = blockIdx.x * blockDim.x + threadIdx.x;
    if (idx >= N_CODES * CODE_SIZE) return;
    int i = idx / CODE_SIZE;
    int k = idx - i * CODE_SIZE;

    float cx = code_x[i];
    float nb = neighbor[i];
    int   ci = cat_idx[i];

    float cat_i     = cat_table[ci * CODE_SIZE + k];
    float centeremb = cx * c_emb[idx];
    float ccode_cat = cx * cat_i;

    // ncat[i] = neighbor[i] * ccat[cat_idx[i]];  ccat[j] = code_x[j]*(code_x[j]*cat_table[cat_idx[j]])
    int   j    = ci;                 // cat_idx[i] < 128, indexes rows of ccat
    float cxj  = code_x[j];
    int   cj   = cat_idx[j];
    float ncat = nb * (cxj * (cxj * cat_table[cj * CODE_SIZE + k]));

    float neighemb = nb * n_emb[idx];

    float* xr = X + (size_t)i * NX;
    xr[k]      = centeremb;
    xr[28 + k] = ccode_cat;
    xr[56 + k] = neighemb;
    xr[84 + k] = ncat;
}

// ---------------------------------------------------------------------------
// Phase 2: Y[4880][112] = adj[4880][4880] @ X[4880][112], f32 WMMA 16x16x4.
// One wave (32 threads) per 16x16 output tile. grid = (305, 7).
// A frag: lane holds adj[mbase+(l&15)][k + 2*(l>>4) + {0,1}]  (float2)
// B frag: lane holds X[k + 2*(l>>4) + {0,1}][nbase+(l&15)]
// C/D:    vgpr v, lane l -> row mbase + v + 8*(l>>4), col nbase + (l&15)
// ---------------------------------------------------------------------------
__global__ __launch_bounds__(32) void gemm_adjX(const float* __restrict__ adj,
                                                const float* __restrict__ X,
                                                float* __restrict__ Y) {
    const int mt   = blockIdx.x;          // 0..304
    const int nt   = blockIdx.y;          // 0..6
    const int lane = threadIdx.x;         // 0..31
    const int half = lane >> 4;           // 0 or 1 -> K sub-pair
    const int l16  = lane & 15;

    const int arow = mt * 16 + l16;
    const int bcol = nt * 16 + l16;

    const float* arowp = adj + (size_t)arow * N_CODES + 2 * half;  // + k
    const float* bp    = X + (size_t)(2 * half) * NX + bcol;       // + k*NX

    v8f acc = {};

    for (int k = 0; k < N_CODES; k += 16) {
        // pull the next chunk of this adj row toward the caches
        __builtin_prefetch(arowp + k + 128, 0, 3);
#pragma unroll
        for (int kk = 0; kk < 16; kk += 4) {
            const int kc = k + kk;
            v2f a = *(const v2f*)(arowp + kc);
            v2f bfrag;
            bfrag.x = bp[(size_t)kc * NX];
            bfrag.y = bp[(size_t)kc * NX + NX];
            acc = __builtin_amdgcn_wmma_f32_16x16x4_f32(
                false, a, false, bfrag, (short)0, acc, false, false);
        }
    }

    float* yp = Y + (size_t)(mt * 16 + half * 8) * NX + bcol;
#pragma unroll
    for (int v = 0; v < 8; ++v) yp[(size_t)v * NX] = acc[v];
}

// ---------------------------------------------------------------------------
// Phase 3a: Z[9760][56]:
//   Z[i]        = center_full[i] + cx*(adj_center[i] + adj_neigh[i])
//   Z[4880 + i] = neigh_full[i]  + nb*(adj_neigh[i]  + adj_center[i])
// One thread per (i, k), k in [0,56).
// ---------------------------------------------------------------------------
__global__ void build_Z(const float* __restrict__ code_x,
                        const float* __restrict__ neighbor,
                        const float* __restrict__ X,
                        const float* __restrict__ Y,
                        float* __restrict__ Z) {
    int idx = blockIdx.x * blockDim.x + threadIdx.x;
    if (idx >= N_CODES * NZ) return;
    int i = idx / NZ;
    int k = idx - i * NZ;

    float ac = Y[(size_t)i * NX + k];        // adj_center col k
    float an = Y[(size_t)i * NX + 56 + k];   // adj_neigh  col k
    float cf = X[(size_t)i * NX + k];        // center_full
    float nf = X[(size_t)i * NX + 56 + k];   // neigh_full
    float cx = code_x[i];
    float nb = neighbor[i];

    Z[(size_t)i * NZ + k]             = cf + cx * ac + cx * an;
    Z[(size_t)(N_CODES + i) * NZ + k] = nf + nb * an + nb * ac;
}

// ---------------------------------------------------------------------------
// Phase 3b: out[9760][32] = leaky_relu(Z[9760][56] @ W[56][32] + b), WMMA f32.
// Rows 0..4879 -> co, rows 4880..9759 -> no; d_out is co||no so row r maps
// directly to out + r*32. grid = (610, 2), one wave per 16x16 tile.
// ---------------------------------------------------------------------------
__global__ __launch_bounds__(32) void gemm_ZW(const float* __restrict__ Z,
                                              const float* __restrict__ W,
                                              const float* __restrict__ b,
                                              float* __restrict__ out) {
    const int mt   = blockIdx.x;          // 0..609
    const int nt   = blockIdx.y;          // 0..1
    const int lane = threadIdx.x;
    const int half = lane >> 4;
    const int l16  = lane & 15;

    const float* zrow = Z + (size_t)(mt * 16 + l16) * NZ + 2 * half;
    const float* wp   = W + (size_t)(2 * half) * GRAPH_SIZE + nt * 16 + l16;

    v8f acc = {};
#pragma unroll
    for (int k = 0; k < NZ; k += 4) {
        v2f a = *(const v2f*)(zrow + k);
        v2f bfrag;
        bfrag.x = wp[(size_t)k * GRAPH_SIZE];
        bfrag.y = wp[(size_t)k * GRAPH_SIZE + GRAPH_SIZE];
        acc = __builtin_amdgcn_wmma_f32_16x16x4_f32(
            false, a, false, bfrag, (short)0, acc, false, false);
    }

    const int col  = nt * 16 + l16;
    const float bias = b[col];
    float* op = out + (size_t)(mt * 16 + half * 8) * GRAPH_SIZE + col;
#pragma unroll
    for (int v = 0; v < 8; ++v) {
        float x = acc[v] + bias;
        op[(size_t)v * GRAPH_SIZE] = (x > 0.0f) ? x : 0.01f * x;
    }
}

// ---------------------------------------------------------------------------
extern "C" void kernel_launch(void* const* d_in, const int* in_sizes, int n_in,
                              void* d_out, int out_size, void* d_ws, size_t ws_size,
                              hipStream_t stream) {
    const float* code_x    = (const float*)d_in[0];
    const float* neighbor  = (const float*)d_in[1];
    const float* c_emb     = (const float*)d_in[2];
    const float* n_emb     = (const float*)d_in[3];
    const float* cat_table = (const float*)d_in[4];
    const int*   cat_idx   = (const int*)d_in[5];
    const float* adj       = (const float*)d_in[6];
    const float* W         = (const float*)d_in[7];
    const float* b         = (const float*)d_in[8];
    float* out = (float*)d_out;

    float* X = (float*)d_ws;                       // 4880*112 f32
    float* Y = X + (size_t)N_CODES * NX;           // 4880*112 f32
    float* Z = Y + (size_t)N_CODES * NX;           // 9760*56  f32

    {
        int n = N_CODES * CODE_SIZE;
        build_X<<<(n + 255) / 256, 256, 0, stream>>>(code_x, neighbor, c_emb,
                                                     n_emb, cat_table, cat_idx, X);
    }
    {
        dim3 grid(N_CODES / 16, NX / 16);          // (305, 7)
        gemm_adjX<<<grid, 32, 0, stream>>>(adj, X, Y);
    }
    {
        int n = N_CODES * NZ;
        build_Z<<<(n + 255) / 256, 256, 0, stream>>>(code_x, neighbor, X, Y, Z);
    }
    {
        dim3 grid((2 * N_CODES) / 16, GRAPH_SIZE / 16);  // (610, 2)
        gemm_ZW<<<grid, 32, 0, stream>>>(Z, W, b, out);
    }
}